// EGENet_88802743812424
// MI455X (gfx1250) — compile-verified
//
#include <hip/hip_runtime.h>

typedef __attribute__((ext_vector_type(2))) float v2f;
typedef __attribute__((ext_vector_type(8))) float v8f;

#define B_    4
#define C_    256
#define H_    128
#define W_    128
#define HW_   (H_*W_)
#define NH_   16
#define HD_   16
#define NTOK  64
#define SCALE 0.25f         // hd^-0.5 = 16^-0.5
#define EPS_  1e-5f

__device__ __forceinline__ v8f wmma4(v2f a, v2f b, v8f c) {
  // D = A(16x4 f32) * B(4x16 f32) + C(16x16 f32)
  return __builtin_amdgcn_wmma_f32_16x16x4_f32(false, a, false, b, (short)0, c,
                                               false, false);
}

// Async copy of 16 contiguous bytes global -> LDS, tracked by ASYNCcnt.
__device__ __forceinline__ void async_b128(unsigned lds_off, const float* gp) {
  asm volatile("global_load_async_to_lds_b128 %0, %1, off"
               :: "v"(lds_off), "v"(gp) : "memory");
}
__device__ __forceinline__ void async_wait0() {
  asm volatile("s_wait_asynccnt 0x0" ::: "memory");
}
__device__ __forceinline__ unsigned lds_addr(const void* p) {
  return (unsigned)(unsigned long long)p;   // low 32 bits = LDS byte offset
}

// ---------------------------------------------------------------------------
// Kernel 1: QKV 1x1 conv as WMMA GEMM. out[oc,p] = sum_c W[oc,c] * X[c,p]
// Block = 4 waves sharing one 16-pixel panel; X panel (256x16 f32, 16 KB) is
// async-staged to LDS once, each wave computes a different 16-row tile.
// Result scattered into q/k/v buffers laid out [window][head][token][dim].
// ---------------------------------------------------------------------------
__global__ __launch_bounds__(128) void k_qkv(const float* __restrict__ x,
                                             const float* __restrict__ wq,
                                             float* __restrict__ qb,
                                             float* __restrict__ kb,
                                             float* __restrict__ vb) {
  __shared__ float sX[C_ * 16];              // [channel][pixel], 16 KB
  const int t    = threadIdx.x;
  const int wave = t >> 5;
  const int lane = t & 31;
  const int sub  = lane >> 4;                // K phase / row+8 select
  const int lr   = lane & 15;

  int blk = blockIdx.x;                      // 4 * 1024 * 12 = 49152
  const int b  = blk / (1024 * 12);
  int rem      = blk - b * (1024 * 12);
  const int nt = rem / 12;                   // pixel tile
  const int mg = rem - nt * 12;              // M group
  const int m0 = (mg * 4 + wave) << 4;       // oc tile base (48 tiles)
  const int p0 = nt << 4;

  const float* X = x + (size_t)b * (C_ * HW_);

  // ---- async stage: 256 rows x 64 B, b128 per lane, 8 per thread ----
  {
    const int colg = (t & 3) * 4;
    const int row0 = t >> 2;                 // 0..31
#pragma unroll
    for (int i = 0; i < 8; ++i) {
      int c = i * 32 + row0;
      async_b128(lds_addr(&sX[c * 16 + colg]), X + c * HW_ + p0 + colg);
    }
    async_wait0();
  }
  __syncthreads();

  const float* wp = wq + (m0 + lr) * C_ + 2 * sub;
  const float* bp = &sX[(2 * sub) * 16 + lr];
  v8f acc = {};
  for (int k0 = 0; k0 < C_; k0 += 4) {
    if (k0 + 16 < C_) __builtin_prefetch(wp + k0 + 16, 0, 1);
    v2f a = *(const v2f*)(wp + k0);          // A[M=m0+lr][k0+2*sub .. +1]
    v2f bb;
    bb.x = bp[k0 * 16];                      // B[k0+2*sub][p0+lr] from LDS
    bb.y = bp[(k0 + 1) * 16];
    acc = wmma4(a, bb, acc);
  }

  // scatter to window layout
  const int p   = p0 + lr;
  const int y   = p >> 7, xx = p & 127;
  const int win = b * 256 + (y >> 3) * 16 + (xx >> 3);
  const int n   = ((y & 7) << 3) + (xx & 7);
#pragma unroll
  for (int r = 0; r < 8; ++r) {
    int M   = m0 + r + 8 * sub;              // global oc in [0,768)
    int sel = M >> 8;                        // 0=q 1=k 2=v
    int c   = M & 255;
    int h   = c >> 4, d = c & 15;
    float* dst = (sel == 0) ? qb : ((sel == 1) ? kb : vb);
    dst[(((size_t)win * NH_ + h) * NTOK + n) * HD_ + d] = acc[r];
  }
}

// ---------------------------------------------------------------------------
// Kernel 2: grouped 3x3 conv (groups = NH) + eval BatchNorm.
// ---------------------------------------------------------------------------
__global__ __launch_bounds__(256) void k_local(const float* __restrict__ x,
                                               const float* __restrict__ wl,
                                               const float* __restrict__ g,
                                               const float* __restrict__ bta,
                                               const float* __restrict__ mu,
                                               const float* __restrict__ var,
                                               float* __restrict__ out) {
  int idx = blockIdx.x * 256 + threadIdx.x;  // B*C*HW
  int p  = idx & (HW_ - 1);
  int bc = idx >> 14;
  int c  = bc & 255;
  int b  = bc >> 8;
  int y  = p >> 7, xx = p & 127;
  int grp = c >> 4;
  const float* xin = x + ((size_t)b * C_ + grp * 16) * HW_;
  const float* w   = wl + c * 16 * 9;
  float s = 0.f;
  for (int ic = 0; ic < 16; ++ic) {
    const float* xc = xin + ic * HW_;
    const float* wc = w + ic * 9;
#pragma unroll
    for (int ky = 0; ky < 3; ++ky) {
      int iy = y + ky - 1;
      if (iy < 0 || iy >= H_) continue;
#pragma unroll
      for (int kx = 0; kx < 3; ++kx) {
        int ix = xx + kx - 1;
        if (ix < 0 || ix >= W_) continue;
        s += xc[iy * W_ + ix] * wc[ky * 3 + kx];
      }
    }
  }
  float sc = g[c] * rsqrtf(var[c] + EPS_);
  out[idx] = (s - mu[c]) * sc + bta[c];
}

// ---------------------------------------------------------------------------
// Kernel 3: window attention. One block (4 waves) per (window, head); wave m
// owns rows [16m,16m+16) of the 64x64 score matrix. K and V (4 KB each) are
// shared by all 4 waves -> async-staged to LDS once.
//   GEMM1: Q(64x16) * K^T(16x64)  (B-frags = ds_load_b64 from sK)
//   bias + softmax via shfl_xor row reductions
//   attn staged to LDS (C-layout -> A-layout), GEMM2: attn(16x64)*V(64x16)
// ---------------------------------------------------------------------------
__global__ __launch_bounds__(128) void k_attn(const float* __restrict__ qb,
                                              const float* __restrict__ kb,
                                              const float* __restrict__ vb,
                                              const float* __restrict__ rel,
                                              float* __restrict__ ob) {
  __shared__ float sK[NTOK * HD_];           // 4 KB
  __shared__ float sV[NTOK * HD_];           // 4 KB
  __shared__ float sAttn[4 * 16 * 64];       // 16 KB
  const int t    = threadIdx.x;
  const int wave = t >> 5;
  const int lane = t & 31;
  const int sub  = lane >> 4;
  const int lr   = lane & 15;

  const int task = blockIdx.x;               // window*NH + head
  const int h    = task & (NH_ - 1);
  const int win  = task >> 4;
  const int b    = win >> 8;
  const int hh   = (win >> 4) & 15;
  const int ww   = win & 15;

  const float* Q = qb + (size_t)task * (NTOK * HD_);
  const float* K = kb + (size_t)task * (NTOK * HD_);
  const float* V = vb + (size_t)task * (NTOK * HD_);
  const int m = wave;                        // row-block of score matrix

  // ---- async stage K and V: 1024 floats each = 256 b128, 2/thread each ----
#pragma unroll
  for (int i = 0; i < 2; ++i) {
    int e = (i * 128 + t) * 4;               // float offset
    async_b128(lds_addr(&sK[e]), K + e);
    async_b128(lds_addr(&sV[e]), V + e);
  }
  async_wait0();
  __syncthreads();

  // A fragments for Q rows (K = 16 -> 4 k-steps), contiguous float2 loads
  v2f afr[4];
#pragma unroll
  for (int kk = 0; kk < 4; ++kk)
    afr[kk] = *(const v2f*)(Q + (m * 16 + lr) * HD_ + 4 * kk + 2 * sub);

  v8f acc[4];
#pragma unroll
  for (int n0 = 0; n0 < 4; ++n0) {
    v8f a = {};
#pragma unroll
    for (int kk = 0; kk < 4; ++kk) {
      // B = K^T: lane reads K[tok=n0*16+lr][d=4kk+2*sub .. +1] from LDS
      v2f bfr = *(const v2f*)(&sK[(n0 * 16 + lr) * HD_ + 4 * kk + 2 * sub]);
      a = wmma4(afr[kk], bfr, a);
    }
    acc[n0] = a;
  }

  // scale + relative-position bias (rel_index computed analytically)
#pragma unroll
  for (int n0 = 0; n0 < 4; ++n0) {
#pragma unroll
    for (int r = 0; r < 8; ++r) {
      int i = m * 16 + r + 8 * sub;          // query token
      int j = n0 * 16 + lr;                  // key token
      int yi = i >> 3, xi = i & 7, yj = j >> 3, xj = j & 7;
      int ridx = (yi - yj + 7) * 15 + (xi - xj + 7);
      acc[n0][r] = acc[n0][r] * SCALE + rel[ridx * NH_ + h];
    }
  }

  // softmax over each row (64 cols = 4 accumulators x 16 lanes in one half)
#pragma unroll
  for (int r = 0; r < 8; ++r) {
    float mx = fmaxf(fmaxf(acc[0][r], acc[1][r]), fmaxf(acc[2][r], acc[3][r]));
#pragma unroll
    for (int mk = 1; mk < 16; mk <<= 1)
      mx = fmaxf(mx, __shfl_xor(mx, mk, 32));
    float sum = 0.f;
#pragma unroll
    for (int n0 = 0; n0 < 4; ++n0) {
      float e = __expf(acc[n0][r] - mx);
      acc[n0][r] = e;
      sum += e;
    }
#pragma unroll
    for (int mk = 1; mk < 16; mk <<= 1)
      sum += __shfl_xor(sum, mk, 32);
    float inv = 1.f / sum;
#pragma unroll
    for (int n0 = 0; n0 < 4; ++n0) acc[n0][r] *= inv;
  }

  // stage probabilities through LDS: C-layout -> row-major 16x64
  float* sw = sAttn + wave * 16 * 64;
#pragma unroll
  for (int n0 = 0; n0 < 4; ++n0)
#pragma unroll
    for (int r = 0; r < 8; ++r)
      sw[(r + 8 * sub) * 64 + n0 * 16 + lr] = acc[n0][r];
  __syncthreads();

  // GEMM2: o(16x16) = attn(16x64) * V(64x16), 16 k-steps of 4
  v8f oacc = {};
#pragma unroll
  for (int ks = 0; ks < 16; ++ks) {
    int k0 = ks * 4;
    v2f afr2 = *(const v2f*)(sw + lr * 64 + k0 + 2 * sub);   // ds_load_b64
    v2f bfr;
    bfr.x = sV[(k0 + 2 * sub) * HD_ + lr];                   // B[tok][d]
    bfr.y = sV[(k0 + 1 + 2 * sub) * HD_ + lr];
    oacc = wmma4(afr2, bfr, oacc);
  }

  // store NCHW
#pragma unroll
  for (int r = 0; r < 8; ++r) {
    int tok = m * 16 + r + 8 * sub;
    int c   = h * HD_ + lr;
    int y   = hh * 8 + (tok >> 3);
    int xx  = ww * 8 + (tok & 7);
    ob[((size_t)(b * C_ + c)) * HW_ + y * W_ + xx] = oacc[r];
  }
}

// ---------------------------------------------------------------------------
// Kernel 4: (o + local) -> reflect-pad(0,1) -> depthwise 3x3 (zero pad) -> BN
// Only the 128x128 crop region is ever needed. coord 128 reflects to 126.
// ---------------------------------------------------------------------------
__device__ __forceinline__ int refl(int v) { return v == 128 ? 126 : v; }

__global__ __launch_bounds__(256) void k_dwbn(const float* __restrict__ o,
                                              const float* __restrict__ loc,
                                              const float* __restrict__ wd,
                                              const float* __restrict__ g,
                                              const float* __restrict__ bta,
                                              const float* __restrict__ mu,
                                              const float* __restrict__ var,
                                              float* __restrict__ out) {
  int idx = blockIdx.x * 256 + threadIdx.x;
  int p  = idx & (HW_ - 1);
  int bc = idx >> 14;
  int c  = bc & 255;
  int y  = p >> 7, xx = p & 127;
  const float* oc_ = o + (size_t)bc * HW_;
  const float* lc  = loc + (size_t)bc * HW_;
  const float* w   = wd + c * 9;
  float s = 0.f;
#pragma unroll
  for (int ky = 0; ky < 3; ++ky) {
    int iy = y + ky - 1;
    if (iy < 0) continue;                    // conv zero pad
    iy = refl(iy);                           // reflect pad of 129-wide input
#pragma unroll
    for (int kx = 0; kx < 3; ++kx) {
      int ix = xx + kx - 1;
      if (ix < 0) continue;
      ix = refl(ix);
      int q = iy * W_ + ix;
      s += (oc_[q] + lc[q]) * w[ky * 3 + kx];
    }
  }
  float sc = g[c] * rsqrtf(var[c] + EPS_);
  out[idx] = (s - mu[c]) * sc + bta[c];
}

// ---------------------------------------------------------------------------
// Kernel 5: pointwise 1x1 conv as WMMA GEMM (256 x 256 x 65536) -> d_out.
// Same LDS-panel structure as k_qkv (16 M-tiles -> 4 groups of 4 waves).
// ---------------------------------------------------------------------------
__global__ __launch_bounds__(128) void k_pw(const float* __restrict__ tin,
                                            const float* __restrict__ wpw,
                                            float* __restrict__ out) {
  __shared__ float sX[C_ * 16];              // 16 KB
  const int t    = threadIdx.x;
  const int wave = t >> 5;
  const int lane = t & 31;
  const int sub  = lane >> 4;
  const int lr   = lane & 15;

  int blk = blockIdx.x;                      // 4 * 1024 * 4 = 16384
  const int b  = blk / (1024 * 4);
  int rem      = blk - b * (1024 * 4);
  const int nt = rem >> 2;
  const int mg = rem & 3;
  const int m0 = (mg * 4 + wave) << 4;
  const int p0 = nt << 4;

  const float* X = tin + (size_t)b * (C_ * HW_);
  {
    const int colg = (t & 3) * 4;
    const int row0 = t >> 2;
#pragma unroll
    for (int i = 0; i < 8; ++i) {
      int c = i * 32 + row0;
      async_b128(lds_addr(&sX[c * 16 + colg]), X + c * HW_ + p0 + colg);
    }
    async_wait0();
  }
  __syncthreads();

  const float* wp = wpw + (m0 + lr) * C_ + 2 * sub;
  const float* bp = &sX[(2 * sub) * 16 + lr];
  v8f acc = {};
  for (int k0 = 0; k0 < C_; k0 += 4) {
    if (k0 + 16 < C_) __builtin_prefetch(wp + k0 + 16, 0, 1);
    v2f a = *(const v2f*)(wp + k0);
    v2f bb;
    bb.x = bp[k0 * 16];
    bb.y = bp[(k0 + 1) * 16];
    acc = wmma4(a, bb, acc);
  }

  float* O = out + (size_t)b * C_ * HW_;
#pragma unroll
  for (int r = 0; r < 8; ++r) {
    int M = m0 + r + 8 * sub;
    O[(size_t)M * HW_ + p0 + lr] = acc[r];
  }
}

// ---------------------------------------------------------------------------
extern "C" void kernel_launch(void* const* d_in, const int* in_sizes, int n_in,
                              void* d_out, int out_size, void* d_ws,
                              size_t ws_size, hipStream_t stream) {
  const float* x       = (const float*)d_in[0];
  const float* w_qkv   = (const float*)d_in[1];
  const float* w_local = (const float*)d_in[2];
  const float* bn_l_g  = (const float*)d_in[3];
  const float* bn_l_b  = (const float*)d_in[4];
  const float* bn_l_m  = (const float*)d_in[5];
  const float* bn_l_v  = (const float*)d_in[6];
  const float* rel     = (const float*)d_in[7];
  const float* w_dw    = (const float*)d_in[8];
  const float* bn_p_g  = (const float*)d_in[9];
  const float* bn_p_b  = (const float*)d_in[10];
  const float* bn_p_m  = (const float*)d_in[11];
  const float* bn_p_v  = (const float*)d_in[12];
  const float* w_pw    = (const float*)d_in[13];
  float* out = (float*)d_out;
  (void)in_sizes; (void)n_in; (void)out_size; (void)ws_size;

  const size_t SZ = (size_t)B_ * C_ * HW_;   // 16,777,216 elements
  float* qb = (float*)d_ws;                  // [win][head][tok][dim]
  float* kb = qb + SZ;
  float* vb = kb + SZ;
  float* lb = vb + SZ;                       // local branch (NCHW)
  float* ob = lb + SZ;                       // attention output (NCHW)
  float* tb = qb;                            // dw+bn output, reuses q region

  k_qkv  <<<49152, 128, 0, stream>>>(x, w_qkv, qb, kb, vb);
  k_local<<<65536, 256, 0, stream>>>(x, w_local, bn_l_g, bn_l_b, bn_l_m,
                                     bn_l_v, lb);
  k_attn <<<16384, 128, 0, stream>>>(qb, kb, vb, rel, ob);
  k_dwbn <<<65536, 256, 0, stream>>>(ob, lb, w_dw, bn_p_g, bn_p_b, bn_p_m,
                                     bn_p_v, tb);
  k_pw   <<<16384, 128, 0, stream>>>(tb, w_pw, out);
}